// SAGEwithJK_40020505264513
// MI455X (gfx1250) — compile-verified
//
#include <hip/hip_runtime.h>

// SAGE (mean aggr) x3 + JumpingKnowledge max, fp32 end-to-end.
// N=50000 nodes, E=800000 edges, dims 128 -> 256 -> 256 -> 256.
// Roofline: GEMMs are ~33 GFLOP (trivial); the 3 gather/scatter passes move
// ~2.4GB of L2-resident rows and do ~512M f32 atomic RMWs -> that is the
// bottleneck, so: fp32 precision kept, V_WMMA_F32_16X16X4_F32 for matmuls,
// TDM tensor_load_to_lds for A-tile staging, global_atomic_add_f32 scatter.

#define NN 50000
#define NE 800000
#define HID 256

typedef __attribute__((ext_vector_type(2))) float v2f;
typedef __attribute__((ext_vector_type(8))) float v8f;
typedef __attribute__((ext_vector_type(4))) unsigned int u32x4;
typedef __attribute__((ext_vector_type(8))) int i32x8;
typedef __attribute__((ext_vector_type(4))) int i32x4;

// ---- hardware f32 global atomic add (no-return form, STOREcnt) ----
__device__ __forceinline__ void atom_add_f32(float* p, float v) {
  asm volatile("global_atomic_add_f32 %0, %1, off" ::"v"(p), "v"(v) : "memory");
}

// ---- TDM: DMA a 16 x K f32 tile (row stride K) from global into LDS ----
// D# bit layout per CDNA5 ISA 8.3/8.4. 2D tensor -> groups 2/3 zero.
template <int K>
__device__ __forceinline__ void tdm_load_tile16(const float* gsrc,
                                                unsigned lds_byte_off) {
  unsigned long long ga = (unsigned long long)(uintptr_t)gsrc;
  u32x4 g0;
  g0[0] = 1u;                                  // count=1, user descriptor
  g0[1] = lds_byte_off;                        // lds_addr (bytes)
  g0[2] = (unsigned)(ga & 0xFFFFFFFFu);        // global_addr[31:0]
  g0[3] = (unsigned)((ga >> 32) & 0x01FFFFFFu) // global_addr[56:32]
          | (2u << 30);                        // type=2 ("image")
  i32x8 g1;
  g1[0] = 0x00020000;                          // data_size=2 (4 bytes)
  g1[1] = (int)((unsigned)(K & 0xFFFF) << 16); // tensor_dim0[15:0]
  g1[2] = (int)(((unsigned)K >> 16) | ((unsigned)(NN & 0xFFFF) << 16)); // dim0 hi | tensor_dim1 lo
  g1[3] = (int)(((unsigned)NN >> 16) | ((unsigned)K << 16)); // tensor_dim1 hi | tile_dim0=K
  g1[4] = 16;                                  // tile_dim1=16 rows (tile_dim2=0)
  g1[5] = K;                                   // tensor_dim0_stride[31:0] = K elems
  g1[6] = 0;
  g1[7] = 0;
  i32x4 z4 = {0, 0, 0, 0};
  i32x8 z8 = {0, 0, 0, 0, 0, 0, 0, 0};
  __builtin_amdgcn_tensor_load_to_lds(g0, g1, z4, z4, z8, 0);
}

__global__ void zero_kernel(float4* __restrict__ p, int n4) {
  int i = blockIdx.x * blockDim.x + threadIdx.x;
  if (i < n4) p[i] = make_float4(0.f, 0.f, 0.f, 0.f);
}

__global__ void deg_count_kernel(const int* __restrict__ edges, unsigned* deg) {
  int e = blockIdx.x * blockDim.x + threadIdx.x;
  if (e < NE) {
    int dst = edges[e];  // edge_index[0] = dst
    __hip_atomic_fetch_add(&deg[dst], 1u, __ATOMIC_RELAXED, __HIP_MEMORY_SCOPE_AGENT);
  }
}

// In place: buffer holds u32 counts, becomes float 1/max(deg,1).
__global__ void deg_inv_kernel(float* buf) {
  int i = blockIdx.x * blockDim.x + threadIdx.x;
  if (i < NN) {
    unsigned d = ((const unsigned*)buf)[i];
    buf[i] = 1.0f / fmaxf((float)d, 1.0f);
  }
}

// One wave per edge: agg[dst, :] += h[src, :] * inv_deg[dst]
// float4 gathers (rows are L2-resident) + global_atomic_add_f32 scatter.
template <int D>
__global__ __launch_bounds__(256) void scatter_mean_kernel(
    const float* __restrict__ h, const int* __restrict__ edges,
    const float* __restrict__ inv_deg, float* __restrict__ agg) {
  const int lane = threadIdx.x & 31;
  const int wave = threadIdx.x >> 5;
  const int e = blockIdx.x * 8 + wave;
  if (e >= NE) return;
  const int dst = edges[e];       // edge_index[0]
  const int src = edges[NE + e];  // edge_index[1]
  const float w = inv_deg[dst];
  const float4* __restrict__ srow = (const float4*)(h + (size_t)src * D);
  float* drow = agg + (size_t)dst * D;
#pragma unroll
  for (int i = 0; i < D / 128; ++i) {
    const int c = i * 32 + lane;  // float4 index within row
    float4 v = srow[c];
    const int f = c * 4;
    atom_add_f32(drow + f + 0, v.x * w);
    atom_add_f32(drow + f + 1, v.y * w);
    atom_add_f32(drow + f + 2, v.z * w);
    atom_add_f32(drow + f + 3, v.w * w);
  }
}

// out[m, n] = act( [agg|hin](m, 0:2K) @ [[Wl],[Wr]] + bl ), fused JK-max.
// Block = 256 thr = 8 waves; block covers 16 rows x 128 cols; each wave owns a
// 16x16 tile via V_WMMA_F32_16X16X4_F32. The two 16xK A-tiles (agg, hin) are
// DMA'd into LDS by the Tensor Data Mover (wave 0 issues both descriptors).
template <int K, bool RELU, bool FIRST, bool STORE_H>
__global__ __launch_bounds__(256) void sage_gemm_kernel(
    const float* __restrict__ agg, const float* __restrict__ hin,
    const float* __restrict__ Wl, const float* __restrict__ bl,
    const float* __restrict__ Wr, float* __restrict__ hout,
    float* __restrict__ outmax) {
  __shared__ float smA[32 * K];  // [0,16K): agg tile, [16K,32K): hin tile
  const int tid = threadIdx.x;
  const int lane = tid & 31;
  const int wave = tid >> 5;
  const int m0 = blockIdx.x * 16;
  const int n0 = blockIdx.y * 128 + wave * 16;

  if (wave == 0) {  // uniform branch: one wave drives the TDM
    const unsigned lds0 = (unsigned)(uintptr_t)&smA[0];
    tdm_load_tile16<K>(agg + (size_t)m0 * K, lds0);
    tdm_load_tile16<K>(hin + (size_t)m0 * K, lds0 + 16u * K * 4u);
    __builtin_amdgcn_s_wait_tensorcnt(0);
  }
  __syncthreads();

  // ISA 16x16x4 f32 A layout: lanes 0-15 hold K={k0,k0+1}, lanes 16-31 K={k0+2,k0+3}
  const int mrow = lane & 15;
  const int khalf = (lane >> 4) << 1;  // 0 or 2
  const int col = n0 + mrow;           // B/C/D column = lane%16
  v8f acc = {};
#pragma unroll 4
  for (int k0 = 0; k0 < 2 * K; k0 += 4) {
    const int kb = k0 + khalf;
    const bool fh = (k0 < K);  // uniform per chunk
    const int kw = fh ? kb : (kb - K);
    const float* __restrict__ As = fh ? smA : (smA + 16 * K);
    const float* __restrict__ W = fh ? Wl : Wr;
    v2f a;
    a.x = As[mrow * K + kw];
    a.y = As[mrow * K + kw + 1];
    v2f b;
    b.x = W[(size_t)kw * HID + col];
    b.y = W[(size_t)(kw + 1) * HID + col];
    acc = __builtin_amdgcn_wmma_f32_16x16x4_f32(false, a, false, b, (short)0,
                                                acc, false, false);
  }

  // Epilogue: bias, optional ReLU, store h, fold JK running max into d_out.
  const float bias = bl[col];
  const int rbase = (lane >> 4) << 3;  // C/D: VGPR r -> row r (lanes<16) / r+8
#pragma unroll
  for (int r = 0; r < 8; ++r) {
    float v = acc[r] + bias;
    if (RELU) v = fmaxf(v, 0.0f);
    const size_t idx = (size_t)(m0 + rbase + r) * HID + col;
    if (STORE_H) hout[idx] = v;
    if (FIRST)
      outmax[idx] = v;
    else
      outmax[idx] = fmaxf(outmax[idx], v);
  }
}

extern "C" void kernel_launch(void* const* d_in, const int* in_sizes, int n_in,
                              void* d_out, int out_size, void* d_ws,
                              size_t ws_size, hipStream_t stream) {
  const float* x = (const float*)d_in[0];
  const int* edges = (const int*)d_in[1];  // JAX default x64-off -> int32, [2,E]
  const float* Wl0 = (const float*)d_in[2];
  const float* bl0 = (const float*)d_in[3];
  const float* Wr0 = (const float*)d_in[4];
  const float* Wl1 = (const float*)d_in[5];
  const float* bl1 = (const float*)d_in[6];
  const float* Wr1 = (const float*)d_in[7];
  const float* Wl2 = (const float*)d_in[8];
  const float* bl2 = (const float*)d_in[9];
  const float* Wr2 = (const float*)d_in[10];
  float* out = (float*)d_out;

  char* ws = (char*)d_ws;
  float* inv = (float*)ws;  // N floats (u32 counts, then 1/deg)
  size_t off = ((size_t)NN * 4 + 255) & ~(size_t)255;
  float* agg = (float*)(ws + off);
  off += (size_t)NN * HID * 4;
  float* hA = (float*)(ws + off);
  off += (size_t)NN * HID * 4;
  float* hB = (float*)(ws + off);

  // Degrees (once; graph is fixed across layers).
  zero_kernel<<<(NN / 4 + 255) / 256, 256, 0, stream>>>((float4*)inv, NN / 4);
  deg_count_kernel<<<(NE + 255) / 256, 256, 0, stream>>>(edges, (unsigned*)inv);
  deg_inv_kernel<<<(NN + 255) / 256, 256, 0, stream>>>(inv);

  dim3 ggrid(NN / 16, HID / 128);  // 3125 x 2, exact

  // Layer 0: in=128
  zero_kernel<<<(NN * 128 / 4 + 255) / 256, 256, 0, stream>>>((float4*)agg, NN * 128 / 4);
  scatter_mean_kernel<128><<<(NE + 7) / 8, 256, 0, stream>>>(x, edges, inv, agg);
  sage_gemm_kernel<128, true, true, true><<<ggrid, 256, 0, stream>>>(
      agg, x, Wl0, bl0, Wr0, hA, out);

  // Layer 1: 256 -> 256
  zero_kernel<<<(NN * 256 / 4 + 255) / 256, 256, 0, stream>>>((float4*)agg, NN * 256 / 4);
  scatter_mean_kernel<256><<<(NE + 7) / 8, 256, 0, stream>>>(hA, edges, inv, agg);
  sage_gemm_kernel<256, true, false, true><<<ggrid, 256, 0, stream>>>(
      agg, hA, Wl1, bl1, Wr1, hB, out);

  // Layer 2: 256 -> 256, no relu, no h store, max only
  zero_kernel<<<(NN * 256 / 4 + 255) / 256, 256, 0, stream>>>((float4*)agg, NN * 256 / 4);
  scatter_mean_kernel<256><<<(NE + 7) / 8, 256, 0, stream>>>(hB, edges, inv, agg);
  sage_gemm_kernel<256, false, false, false><<<ggrid, 256, 0, stream>>>(
      agg, hB, Wl2, bl2, Wr2, nullptr, out);
}